// FusedAmaxObsFakeQuantize_80427557585490
// MI455X (gfx1250) — compile-verified
//
#include <hip/hip_runtime.h>
#include <hip/hip_bf16.h>
#include <math.h>

// FusedAmaxObsFakeQuantize for MI455X (gfx1250).
// Pure HBM-streaming kernel: 256 MiB read + 256 MiB written once each ->
// ~23 us roofline at 23.3 TB/s. No matmul structure, so no WMMA; the win
// is all in the memory path: gfx1250 async-to-LDS engine with a 4-deep
// per-wave pipeline (ASYNCcnt), b128 NT stores, and a hierarchical
// wave32 max reduction (shfl butterfly -> ds_max_u32 -> global_atomic_max_u32).

typedef __attribute__((ext_vector_type(4))) float v4f;

#define TPB 256
#define ITERS_PER_THREAD 32
#define PIPE 4              // async pipeline depth (LDS buffers per thread)
#define QMAXF 256.0f

// ---------------- gfx1250 async-load-to-LDS detection ----------------
#ifndef FORCE_DIRECT
#define FORCE_DIRECT 0
#endif

#if defined(__has_builtin)
#if __has_builtin(__builtin_amdgcn_global_load_async_to_lds_b128) && !FORCE_DIRECT
#define HAVE_ASYNC 1
#endif
#endif
#ifndef HAVE_ASYNC
#define HAVE_ASYNC 0
#endif

#if HAVE_ASYNC
// Builtin signature (probe-confirmed): (global v4i*, shared v4i*, imm offset, imm cpol)
typedef int i4v __attribute__((vector_size(16)));
typedef __attribute__((address_space(1))) void  GVoid;
typedef __attribute__((address_space(3))) void  LVoid;
typedef __attribute__((address_space(1))) i4v   Gi4;
typedef __attribute__((address_space(3))) i4v   Li4;

// cpol: TH[2:0]=1 -> TH_LOAD_NT (streamed input is touched exactly once;
// keep it evict-first in the 192MB L2), scope bits 0 (WGP).
#define ASYNC_CPOL 1

__device__ __forceinline__ void async_copy_b128(const v4f* g, v4f* l) {
  // GLOBAL_LOAD_ASYNC_TO_LDS_B128: memory -> LDS, tracked by per-wave ASYNCcnt
  __builtin_amdgcn_global_load_async_to_lds_b128(
      (Gi4*)(GVoid*)g, (Li4*)(LVoid*)l, 0, ASYNC_CPOL);
}

#if __has_builtin(__builtin_amdgcn_s_wait_asynccnt)
#define ASYNC_WAIT_IMM(K) __builtin_amdgcn_s_wait_asynccnt(K)
#else
#define ASYNC_WAIT_IMM(K) asm volatile("s_wait_asynccnt " #K ::: "memory")
#endif

// s_wait_asynccnt needs an immediate; select among the PIPE thresholds.
__device__ __forceinline__ void wait_async_le(int n) {
  switch (n) {
    case 0:  ASYNC_WAIT_IMM(0); break;
    case 1:  ASYNC_WAIT_IMM(1); break;
    case 2:  ASYNC_WAIT_IMM(2); break;
    default: ASYNC_WAIT_IMM(3); break;
  }
}
#endif

// ---------------- scalar prologue: scale factor + history roll ----------------
// ws[0] = sf, ws[1] = 1/sf, ((unsigned*)ws)[2] = running amax bits (zeroed)
__global__ void fq_setup_kernel(const float* __restrict__ h,
                                float* __restrict__ ws,
                                float* __restrict__ out_hist) {
  const float h0 = h[0], h1 = h[1], h2 = h[2], h3 = h[3], h4 = h[4];
  const float amax = fmaxf(fmaxf(fmaxf(h0, h1), fmaxf(h2, h3)), h4);
  float sf = 1.0f;
  if (amax > 0.0f && isfinite(amax)) {
    int e = ilogbf(QMAXF / amax);   // exact floor(log2(v))
    if (e < 0) e = 0;
    sf = ldexpf(1.0f, e);           // exact power of two
  }
  ws[0] = sf;
  ws[1] = 1.0f / sf;                // exact: sf is a power of two >= 1
  ((unsigned int*)ws)[2] = 0u;      // global amax accumulator (bits of non-neg f32)
  // new_history = roll(h,-1).at[0].set(m)  ->  [m, h2, h3, h4, h0]
  out_hist[1] = h2;
  out_hist[2] = h3;
  out_hist[3] = h4;
  out_hist[4] = h0;
}

// ---------------- per-element fake quant ----------------
__device__ __forceinline__ float fq1(float x, float sf, float inv, float& m) {
  float y = __builtin_rintf(x * sf);            // v_rndne_f32: round-half-even (jnp.round)
  y = fminf(fmaxf(y, -QMAXF), QMAXF);           // clip to [-256, 256]
  y *= inv;                                     // exact rescale (power of two)
  m = fmaxf(m, __builtin_fabsf(y));
  return y;
}

__device__ __forceinline__ v4f fq4(v4f v, float sf, float inv, float& m) {
  v4f q;
  q.x = fq1(v.x, sf, inv, m);
  q.y = fq1(v.y, sf, inv, m);
  q.z = fq1(v.z, sf, inv, m);
  q.w = fq1(v.w, sf, inv, m);
  return q;
}

// ---------------- main streaming kernel ----------------
__global__ void __launch_bounds__(TPB)
fq_main_kernel(const v4f* __restrict__ x, v4f* __restrict__ out,
               const float* __restrict__ ws, unsigned int* __restrict__ gmax,
               int n4) {
  const float sf  = ws[0];
  const float inv = ws[1];
  const int tid    = blockIdx.x * TPB + threadIdx.x;
  const int stride = gridDim.x * TPB;
  float m = 0.0f;

#if HAVE_ASYNC
  // 4-deep per-wave async pipeline: each thread owns one 16B slot in each of
  // PIPE buffers. ASYNCcnt is per-wave and async loads retire in order, so
  // s_wait_asynccnt (PIPE-1) in steady state guarantees the oldest transfer
  // has landed while PIPE-1 remain in flight (2 KB/wave of read MLP).
  // No cross-wave barriers: waves only touch their own LDS slots.
  // NOTE: launch keeps iteration counts uniform across lanes for our sizes
  // (n4 divisible by stride), so the wave-scalar waits match per-lane state.
  __shared__ v4f tile[PIPE][TPB];
  int i = tid;
  int slot = 0;

  // Prologue: fill the pipeline.
  #pragma unroll
  for (int k = 0; k < PIPE; ++k) {
    const int ip = tid + k * stride;
    if (ip < n4) async_copy_b128(&x[ip], &tile[k][threadIdx.x]);
  }

  // Steady state: always PIPE transfers outstanding.
  for (; i + PIPE * stride < n4; i += stride) {
    ASYNC_WAIT_IMM(PIPE - 1);                   // oldest transfer retired
    const v4f v = tile[slot][threadIdx.x];      // ds_load_b128
    asm volatile("s_wait_dscnt 0" ::: "memory"); // v is in VGPRs: slot reusable
    async_copy_b128(&x[i + PIPE * stride], &tile[slot][threadIdx.x]);
    const v4f q = fq4(v, sf, inv, m);
    __builtin_nontemporal_store(q, &out[i]);    // global_store_b128 th:NT
    slot = (slot + 1) & (PIPE - 1);
  }

  // Drain: no new issues; outstanding count steps down, so must the waits.
  int remaining = (i < n4) ? ((n4 - i + stride - 1) / stride) : 0;
  for (; i < n4; i += stride) {
    wait_async_le(remaining - 1);
    const v4f v = tile[slot][threadIdx.x];
    const v4f q = fq4(v, sf, inv, m);
    __builtin_nontemporal_store(q, &out[i]);
    slot = (slot + 1) & (PIPE - 1);
    --remaining;
  }
#else
  // Direct fallback: NT b128 loads + software prefetch one grid-stride ahead.
  #pragma unroll 4
  for (int i = tid; i < n4; i += stride) {
    if (i + stride < n4) __builtin_prefetch((const void*)&x[i + stride], 0, 1);
    const v4f v = __builtin_nontemporal_load(&x[i]);   // global_load_b128 th:NT
    const v4f q = fq4(v, sf, inv, m);
    __builtin_nontemporal_store(q, &out[i]);           // global_store_b128 th:NT
  }
#endif

  // ---- hierarchical max reduction: wave32 butterfly -> LDS -> global ----
  #pragma unroll
  for (int off = 16; off > 0; off >>= 1)
    m = fmaxf(m, __shfl_xor(m, off, 32));

  __shared__ unsigned int smax;
  if (threadIdx.x == 0) smax = 0u;
  __syncthreads();
  if ((threadIdx.x & 31) == 0)
    atomicMax(&smax, __float_as_uint(m));       // ds_max_u32 (m >= 0: bits order as u32)
  __syncthreads();
  if (threadIdx.x == 0)
    atomicMax(gmax, smax);                      // global_atomic_max_u32, 1 per block
}

// ---------------- epilogue: publish amax into new_history[0] ----------------
__global__ void fq_finalize_kernel(const unsigned int* __restrict__ gmax,
                                   float* __restrict__ out_hist) {
  out_hist[0] = __uint_as_float(*gmax);
}

extern "C" void kernel_launch(void* const* d_in, const int* in_sizes, int n_in,
                              void* d_out, int out_size, void* d_ws, size_t ws_size,
                              hipStream_t stream) {
  const float* x = (const float*)d_in[0];        // (4, 4096, 4096) f32
  const float* h = (const float*)d_in[1];        // (5,) f32
  float* out = (float*)d_out;                    // [x_q (n) | new_history (5)]
  const int n  = in_sizes[0];
  const int n4 = n >> 2;                         // n is a multiple of 4

  float* wsf = (float*)d_ws;                     // ws[0]=sf ws[1]=1/sf ws[2]=amax bits
  unsigned int* gmax = ((unsigned int*)d_ws) + 2;

  fq_setup_kernel<<<1, 1, 0, stream>>>(h, wsf, out + n);

  int blocks = (n4 + TPB * ITERS_PER_THREAD - 1) / (TPB * ITERS_PER_THREAD);
  if (blocks < 1) blocks = 1;                    // 2048 blocks -> 16384 wave32s
  fq_main_kernel<<<blocks, TPB, 0, stream>>>((const v4f*)x, (v4f*)out, wsf, gmax, n4);

  fq_finalize_kernel<<<1, 1, 0, stream>>>(gmax, out + n);
}